// SSIM_687194768296
// MI455X (gfx1250) — compile-verified
//
#include <hip/hip_runtime.h>
#include <hip/hip_bf16.h>

typedef __attribute__((ext_vector_type(2))) float v2f;
typedef __attribute__((ext_vector_type(8))) float v8f;

#define RAD 5
#define KW 11
#define TILE 16
#define WAVES 8
#define STRIP_COLS 128
#define ROW_TILES 2
#define STRIP_ROWS (TILE * ROW_TILES)          // 32
#define P_ROWS (STRIP_ROWS + 2 * RAD)          // 42
#define P_STRIDE 149                           // 138 valid + zero pad, odd stride vs 64 banks
#define T_STRIDE 33
#define IMG_H 512
#define IMG_W 512
#define C1F 0.0001f
#define C2F 0.0009f

__device__ __forceinline__ v8f wmma4(v2f a, v2f b, v8f c) {
  // D(16x16,f32) = A(16x4,f32) * B(4x16,f32) + C
  return __builtin_amdgcn_wmma_f32_16x16x4_f32(
      /*neg_a=*/false, a, /*neg_b=*/false, b,
      /*c_mod=*/(short)0, c, /*reuse_a=*/false, /*reuse_b=*/false);
}

__global__ __launch_bounds__(256) void ssim_tile_kernel(
    const float* __restrict__ img1, const float* __restrict__ img2,
    const float* __restrict__ window, float* __restrict__ partials) {
  extern __shared__ float lds[];
  float* P1s = lds;                              // [P_ROWS][P_STRIDE]
  float* P2s = P1s + P_ROWS * P_STRIDE;          // [P_ROWS][P_STRIDE]
  float* Ts  = P2s + P_ROWS * P_STRIDE;          // [WAVES][5][TILE][T_STRIDE]
  __shared__ float s_g[16];
  __shared__ float s_wsum[WAVES];

  const int tid = threadIdx.x;

  // Recover 1-D gaussian taps from 2-D window (outer product): g[j] = w2d[5][j] / sqrt(w2d[5][5])
  if (tid < KW) {
    float w55 = window[5 * KW + 5];
    s_g[tid] = window[5 * KW + tid] * rsqrtf(w55);
  }

  // Stage img1/img2 halo patches into LDS (zero-padded boundaries, SAME conv)
  const int r0 = (int)blockIdx.y * STRIP_ROWS - RAD;
  const int c0 = (int)blockIdx.x * STRIP_COLS - RAD;
  const size_t ibase = (size_t)blockIdx.z * (IMG_H * IMG_W);
  for (int idx = tid; idx < P_ROWS * P_STRIDE; idx += 256) {
    int pr = idx / P_STRIDE;
    int pc = idx - pr * P_STRIDE;
    int r = r0 + pr, c = c0 + pc;
    float a = 0.f, b = 0.f;
    if (pc < STRIP_COLS + 2 * RAD && r >= 0 && r < IMG_H && c >= 0 && c < IMG_W) {
      a = img1[ibase + (size_t)r * IMG_W + c];
      b = img2[ibase + (size_t)r * IMG_W + c];
    }
    P1s[idx] = a;
    P2s[idx] = b;
  }
  __syncthreads();

  const int lane = tid & 31;
  const int wave = tid >> 5;
  const int n = lane & 15;   // N (cols) for B/C/D; M (rows) for A
  const int h = lane >> 4;   // lane half selects K pairs
  float* Tw = Ts + wave * (5 * TILE * T_STRIDE);

  // Banded-Toeplitz gaussian operand: band[m or n][k] = g[k-m], K padded 26->28.
  // Identical element mapping for vertical-pass A and horizontal-pass B.
  v2f wb[7];
#pragma unroll
  for (int s = 0; s < 7; ++s) {
    int k0 = 4 * s + 2 * h;
    int k1 = k0 + 1;
    int d0 = k0 - n, d1 = k1 - n;
    wb[s].x = (k0 < 26 && d0 >= 0 && d0 <= 2 * RAD) ? s_g[d0] : 0.f;
    wb[s].y = (k1 < 26 && d1 >= 0 && d1 <= 2 * RAD) ? s_g[d1] : 0.f;
  }

  float lsum = 0.f;

  for (int rt = 0; rt < ROW_TILES; ++rt) {
    const int rbase = rt * TILE;

    // ---- vertical pass: T[i][j] = sum_k g[k] * q[rbase+i+k][wave*16 + j], two 16-col blocks ----
#pragma unroll
    for (int cblk = 0; cblk < 2; ++cblk) {
      const int colB = wave * TILE + cblk * 16 + n;
      v8f a0 = {0,0,0,0,0,0,0,0}, a1 = {0,0,0,0,0,0,0,0}, a2 = {0,0,0,0,0,0,0,0};
      v8f a3 = {0,0,0,0,0,0,0,0}, a4 = {0,0,0,0,0,0,0,0};
#pragma unroll
      for (int s = 0; s < 7; ++s) {
        int j0 = 4 * s + 2 * h;
        int j1 = j0 + 1;
        float p1a = 0.f, p2a = 0.f, p1b = 0.f, p2b = 0.f;
        if (j0 < 26) {
          int o = (rbase + j0) * P_STRIDE + colB;
          p1a = P1s[o]; p2a = P2s[o];
        }
        if (j1 < 26) {
          int o = (rbase + j1) * P_STRIDE + colB;
          p1b = P1s[o]; p2b = P2s[o];
        }
        v2f b;
        b.x = p1a;         b.y = p1b;         a0 = wmma4(wb[s], b, a0);  // i1
        b.x = p2a;         b.y = p2b;         a1 = wmma4(wb[s], b, a1);  // i2
        b.x = p1a * p1a;   b.y = p1b * p1b;   a2 = wmma4(wb[s], b, a2);  // i1*i1
        b.x = p2a * p2a;   b.y = p2b * p2b;   a3 = wmma4(wb[s], b, a3);  // i2*i2
        b.x = p1a * p2a;   b.y = p1b * p2b;   a4 = wmma4(wb[s], b, a4);  // i1*i2
      }
      // D layout -> T in LDS (row M = r + 8*h, col N = n)
#pragma unroll
      for (int r = 0; r < 8; ++r) {
        int row = r + 8 * h;
        int col = cblk * 16 + n;
        Tw[(0 * TILE + row) * T_STRIDE + col] = a0[r];
        Tw[(1 * TILE + row) * T_STRIDE + col] = a1[r];
        Tw[(2 * TILE + row) * T_STRIDE + col] = a2[r];
        Tw[(3 * TILE + row) * T_STRIDE + col] = a3[r];
        Tw[(4 * TILE + row) * T_STRIDE + col] = a4[r];
      }
    }

    // ---- horizontal pass: Out[i][c] = sum_k g[k] * T[i][c+k] ----
    v8f o0 = {0,0,0,0,0,0,0,0}, o1 = {0,0,0,0,0,0,0,0}, o2 = {0,0,0,0,0,0,0,0};
    v8f o3 = {0,0,0,0,0,0,0,0}, o4 = {0,0,0,0,0,0,0,0};
#pragma unroll
    for (int s = 0; s < 7; ++s) {
      int k0 = 4 * s + 2 * h;
      v2f aq;
      aq.x = Tw[(0 * TILE + n) * T_STRIDE + k0];
      aq.y = Tw[(0 * TILE + n) * T_STRIDE + k0 + 1];
      o0 = wmma4(aq, wb[s], o0);
      aq.x = Tw[(1 * TILE + n) * T_STRIDE + k0];
      aq.y = Tw[(1 * TILE + n) * T_STRIDE + k0 + 1];
      o1 = wmma4(aq, wb[s], o1);
      aq.x = Tw[(2 * TILE + n) * T_STRIDE + k0];
      aq.y = Tw[(2 * TILE + n) * T_STRIDE + k0 + 1];
      o2 = wmma4(aq, wb[s], o2);
      aq.x = Tw[(3 * TILE + n) * T_STRIDE + k0];
      aq.y = Tw[(3 * TILE + n) * T_STRIDE + k0 + 1];
      o3 = wmma4(aq, wb[s], o3);
      aq.x = Tw[(4 * TILE + n) * T_STRIDE + k0];
      aq.y = Tw[(4 * TILE + n) * T_STRIDE + k0 + 1];
      o4 = wmma4(aq, wb[s], o4);
    }

    // ---- SSIM map on D layout + local sum ----
#pragma unroll
    for (int r = 0; r < 8; ++r) {
      float mu1 = o0[r], mu2 = o1[r];
      float x11 = o2[r], x22 = o3[r], x12 = o4[r];
      float mu1s = mu1 * mu1, mu2s = mu2 * mu2, m12 = mu1 * mu2;
      float sg1 = x11 - mu1s, sg2 = x22 - mu2s, sg12 = x12 - m12;
      float num = (2.f * m12 + C1F) * (2.f * sg12 + C2F);
      float den = (mu1s + mu2s + C1F) * (sg1 + sg2 + C2F);
      lsum += num / den;
    }
  }

  // deterministic per-workgroup partial
#pragma unroll
  for (int off = 16; off > 0; off >>= 1) lsum += __shfl_xor(lsum, off);
  if (lane == 0) s_wsum[wave] = lsum;
  __syncthreads();
  if (tid == 0) {
    float p = 0.f;
#pragma unroll
    for (int w = 0; w < WAVES; ++w) p += s_wsum[w];
    int wg = ((int)blockIdx.z * gridDim.y + (int)blockIdx.y) * gridDim.x + (int)blockIdx.x;
    partials[wg] = p;
  }
}

__global__ __launch_bounds__(256) void ssim_reduce_kernel(
    const float* __restrict__ partials, int nPart, float invN, float* __restrict__ out) {
  __shared__ float buf[256];
  float s = 0.f;
  for (int i = threadIdx.x; i < nPart; i += 256) s += partials[i];
  buf[threadIdx.x] = s;
  __syncthreads();
  for (int off = 128; off > 0; off >>= 1) {
    if ((int)threadIdx.x < off) buf[threadIdx.x] += buf[threadIdx.x + off];
    __syncthreads();
  }
  if (threadIdx.x == 0) out[0] = buf[0] * invN;
}

extern "C" void kernel_launch(void* const* d_in, const int* in_sizes, int n_in,
                              void* d_out, int out_size, void* d_ws, size_t ws_size,
                              hipStream_t stream) {
  const float* img1 = (const float*)d_in[0];
  const float* img2 = (const float*)d_in[1];
  const float* window = (const float*)d_in[2];
  float* out = (float*)d_out;
  float* partials = (float*)d_ws;

  const int nImg = in_sizes[0] / (IMG_H * IMG_W);  // B*C = 48
  dim3 grid(IMG_W / STRIP_COLS, IMG_H / STRIP_ROWS, nImg);
  const size_t ldsBytes =
      (size_t)(2 * P_ROWS * P_STRIDE + WAVES * 5 * TILE * T_STRIDE) * sizeof(float);

  ssim_tile_kernel<<<grid, 256, ldsBytes, stream>>>(img1, img2, window, partials);

  const int nPart = grid.x * grid.y * grid.z;
  const float invN = 1.0f / ((float)in_sizes[0]);
  ssim_reduce_kernel<<<1, 256, 0, stream>>>(partials, nPart, invN, out);
}